// SimpleGATLayer_83476984365539
// MI455X (gfx1250) — compile-verified
//
#include <hip/hip_runtime.h>
#include <math.h>

// ---------------- problem constants (from reference) ----------------
#define INDIM  256
#define OUTDIM 64
#define HEADS  4

typedef float v2f __attribute__((ext_vector_type(2)));
typedef float v8f __attribute__((ext_vector_type(8)));

// =====================================================================
// Kernel 1: z[head][n][0..63] = (h[n]*norm[n]) @ W[head]  via f32 WMMA
// block = 128 threads (4 waves). Each wave computes a 16x16 tile of z.
// K is staged through LDS in chunks of 64, consumed 4-at-a-time by
// v_wmma_f32_16x16x4_f32.
// =====================================================================
#define KT 64
__global__ __launch_bounds__(128) void gat_gemm_z(
    const float* __restrict__ h, const float* __restrict__ norm,
    const float* __restrict__ W, float* __restrict__ z, int nNodes)
{
    __shared__ float hT[16][KT + 4];   // stride 68: a-frag banks disjoint
    __shared__ float Wl[KT][72];       // stride 72: 2*72 % 64 == 16 -> halves disjoint

    const int head = blockIdx.y;
    const int n0   = blockIdx.x * 16;
    const int tid  = threadIdx.x;
    const int wv   = tid >> 5;         // wave id 0..3 -> 16-col slice of OUTDIM
    const int lane = tid & 31;
    const int half = lane >> 4;        // 0: K pair {k,k+1}, 1: {k+2,k+3}
    const int r16  = lane & 15;        // row (A) / col (B) within tile

    const float* Wh = W + (size_t)head * INDIM * OUTDIM;

    v8f c = {};   // 16x16 f32 accumulator (8 VGPRs)

    for (int k0 = 0; k0 < INDIM; k0 += KT) {
        // stage h tile (16 x 64), norm folded in
        for (int t = tid; t < 16 * KT; t += 128) {
            int r = t >> 6, cix = t & 63;
            int nIdx = n0 + r; if (nIdx >= nNodes) nIdx = nNodes - 1;
            hT[r][cix] = h[(size_t)nIdx * INDIM + k0 + cix] * norm[nIdx];
        }
        // stage W tile (64 x 64)
        for (int t = tid; t < KT * 64; t += 128) {
            int r = t >> 6, cix = t & 63;
            Wl[r][cix] = Wh[(size_t)(k0 + r) * OUTDIM + cix];
        }
        __syncthreads();

        #pragma unroll
        for (int kk = 0; kk < KT; kk += 4) {
            v2f a, b;
            a.x = hT[r16][kk + 2 * half];
            a.y = hT[r16][kk + 2 * half + 1];
            b.x = Wl[kk + 2 * half][wv * 16 + r16];
            b.y = Wl[kk + 2 * half + 1][wv * 16 + r16];
            // D = A(16x4) * B(4x16) + C   -> v_wmma_f32_16x16x4_f32
            c = __builtin_amdgcn_wmma_f32_16x16x4_f32(
                    false, a, false, b, (short)0, c, false, false);
        }
        __syncthreads();
    }

    // store: VGPR v -> row (v + half*8), col r16 of this wave's 16-col slice
    float* zt = z + ((size_t)head * nNodes + n0) * OUTDIM + wv * 16;
    #pragma unroll
    for (int v = 0; v < 8; v++) {
        int row = v + half * 8;
        if (n0 + row < nNodes) zt[(size_t)row * OUTDIM + r16] = c[v];
    }
}

// =====================================================================
// Kernel 2: per (head,node) score halves  s_src = z . a[:64], s_dst = z . a[64:]
// =====================================================================
__global__ void gat_scores(const float* __restrict__ z, const float* __restrict__ a,
                           float* __restrict__ sS, float* __restrict__ sD, int nNodes)
{
    int t = blockIdx.x * blockDim.x + threadIdx.x;
    if (t >= nNodes * HEADS) return;
    int head = t / nNodes;
    int n    = t - head * nNodes;
    const float4* zr = (const float4*)(z + ((size_t)head * nNodes + n) * OUTDIM);
    const float4* as = (const float4*)(a + head * 2 * OUTDIM);
    const float4* ad = (const float4*)(a + head * 2 * OUTDIM + OUTDIM);
    float s0 = 0.f, s1 = 0.f;
    #pragma unroll
    for (int j = 0; j < OUTDIM / 4; j++) {
        float4 v = zr[j], x = as[j], y = ad[j];
        s0 += v.x * x.x + v.y * x.y + v.z * x.z + v.w * x.w;
        s1 += v.x * y.x + v.y * y.y + v.z * y.z + v.w * y.w;
    }
    sS[t] = s0; sD[t] = s1;
}

// =====================================================================
// CSR-by-dst construction (deterministic after per-bucket sort)
// =====================================================================
__global__ void gat_zero_i32(int* __restrict__ p, int n)
{
    int t = blockIdx.x * blockDim.x + threadIdx.x;
    if (t < n) p[t] = 0;
}

__global__ void gat_hist(const int* __restrict__ dst, int* __restrict__ counts, int nE)
{
    int t = blockIdx.x * blockDim.x + threadIdx.x;
    if (t < nE) atomicAdd(&counts[dst[t]], 1);
}

__global__ __launch_bounds__(1024) void gat_scan_excl(
    const int* __restrict__ counts, int* __restrict__ offsets,
    int* __restrict__ cursor, int n)
{
    __shared__ int buf[1024];
    __shared__ int carry;
    int tid = threadIdx.x;
    if (tid == 0) carry = 0;
    __syncthreads();
    for (int base = 0; base < n; base += 1024) {
        int i = base + tid;
        int v = (i < n) ? counts[i] : 0;
        buf[tid] = v;
        __syncthreads();
        for (int off = 1; off < 1024; off <<= 1) {
            int t = (tid >= off) ? buf[tid - off] : 0;
            __syncthreads();
            buf[tid] += t;
            __syncthreads();
        }
        int excl = buf[tid] - v;
        int cbase = carry;
        if (i < n) { offsets[i] = cbase + excl; cursor[i] = cbase + excl; }
        __syncthreads();
        if (tid == 0) carry = cbase + buf[1023];
        __syncthreads();
    }
    if (tid == 0) offsets[n] = carry;
}

__global__ void gat_scatter(const int* __restrict__ dst, int* __restrict__ cursor,
                            int* __restrict__ eid, int nE)
{
    int t = blockIdx.x * blockDim.x + threadIdx.x;
    if (t < nE) {
        int pos = atomicAdd(&cursor[dst[t]], 1);
        eid[pos] = t;
    }
}

// in-place insertion sort of each bucket -> deterministic reduction order
__global__ void gat_sort_buckets(const int* __restrict__ offsets, int* __restrict__ eid, int n)
{
    int t = blockIdx.x * blockDim.x + threadIdx.x;
    if (t >= n) return;
    int b = offsets[t], e = offsets[t + 1];
    for (int i = b + 1; i < e; i++) {
        int key = eid[i];
        int j = i - 1;
        while (j >= b && eid[j] > key) { eid[j + 1] = eid[j]; j--; }
        eid[j + 1] = key;
    }
}

// =====================================================================
// Kernel 3: pull-mode softmax + aggregation. One wave per (head,node);
// lanes own output dims {lane, lane+32}. No float atomics; z gathers hit L2.
// out[n][head*64+d] = h[n][head*64+d] + norm[n] * sum_e alpha_e z[src_e][d]
// =====================================================================
__global__ __launch_bounds__(256) void gat_aggregate(
    const float* __restrict__ hin, const float* __restrict__ norm,
    const float* __restrict__ z, const float* __restrict__ sS,
    const float* __restrict__ sD, const int* __restrict__ srcIdx,
    const int* __restrict__ offsets, const int* __restrict__ eid,
    float* __restrict__ out, int nNodes)
{
    int wid  = blockIdx.x * 8 + (threadIdx.x >> 5);
    int lane = threadIdx.x & 31;
    if (wid >= nNodes * HEADS) return;
    int head = wid / nNodes;
    int n    = wid - head * nNodes;

    int b = offsets[n], e = offsets[n + 1];
    const float* sSh = sS + (size_t)head * nNodes;
    const float* zh  = z + (size_t)head * nNodes * OUTDIM;

    // pass 1: segment max (score = sS[src]+sD[n]; the +sD[n] constant cancels)
    float m = -INFINITY;
    for (int j = b; j < e; j++) m = fmaxf(m, sSh[srcIdx[eid[j]]]);

    // pass 2: unnormalized weighted sum + denominator (den >= 1, no div0)
    float den = 0.f, acc0 = 0.f, acc1 = 0.f;
    for (int j = b; j < e; j++) {
        int s = srcIdx[eid[j]];
        float w = __expf(sSh[s] - m);
        den += w;
        const float* zr = zh + (size_t)s * OUTDIM;
        acc0 += w * zr[lane];
        acc1 += w * zr[lane + 32];
    }
    float nrm = norm[n];
    float r0 = (e > b) ? acc0 / den * nrm : 0.f;
    float r1 = (e > b) ? acc1 / den * nrm : 0.f;

    size_t o = (size_t)n * INDIM + head * OUTDIM;
    out[o + lane]      = hin[o + lane]      + r0;
    out[o + lane + 32] = hin[o + lane + 32] + r1;
}

// =====================================================================
// launch
// =====================================================================
extern "C" void kernel_launch(void* const* d_in, const int* in_sizes, int n_in,
                              void* d_out, int out_size, void* d_ws, size_t ws_size,
                              hipStream_t stream)
{
    const float* h    = (const float*)d_in[0];
    const float* e    = (const float*)d_in[1];
    const float* norm = (const float*)d_in[2];
    const int*   src  = (const int*)d_in[3];
    const int*   dst  = (const int*)d_in[4];
    const float* W    = (const float*)d_in[5];
    const float* a    = (const float*)d_in[6];

    const int nNodes = in_sizes[2];       // norm has N elements
    const int nE     = in_sizes[1];       // e has E elements

    // ---- workspace carving (256B aligned) ----
    size_t off = 0;
    auto carve = [&](size_t bytes) -> void* {
        void* p = (char*)d_ws + off;
        off = (off + bytes + 255) & ~(size_t)255;
        return p;
    };
    float* z       = (float*)carve((size_t)HEADS * nNodes * OUTDIM * sizeof(float));
    float* sS      = (float*)carve((size_t)HEADS * nNodes * sizeof(float));
    float* sD      = (float*)carve((size_t)HEADS * nNodes * sizeof(float));
    int*   counts  = (int*)carve((size_t)nNodes * sizeof(int));
    int*   offsets = (int*)carve((size_t)(nNodes + 1) * sizeof(int));
    int*   cursor  = (int*)carve((size_t)nNodes * sizeof(int));
    int*   eid     = (int*)carve((size_t)nE * sizeof(int));
    (void)ws_size;

    float* out = (float*)d_out;

    // 1) z = (h*norm) @ W per head   (WMMA)
    dim3 gGemm((nNodes + 15) / 16, HEADS);
    gat_gemm_z<<<gGemm, 128, 0, stream>>>(h, norm, W, z, nNodes);

    // 2) score halves
    int nPairs = nNodes * HEADS;
    gat_scores<<<(nPairs + 255) / 256, 256, 0, stream>>>(z, a, sS, sD, nNodes);

    // 3) CSR by dst
    gat_zero_i32<<<(nNodes + 255) / 256, 256, 0, stream>>>(counts, nNodes);
    gat_hist<<<(nE + 255) / 256, 256, 0, stream>>>(dst, counts, nE);
    gat_scan_excl<<<1, 1024, 0, stream>>>(counts, offsets, cursor, nNodes);
    gat_scatter<<<(nE + 255) / 256, 256, 0, stream>>>(dst, cursor, eid, nE);
    gat_sort_buckets<<<(nNodes + 255) / 256, 256, 0, stream>>>(offsets, eid, nNodes);

    // 4) softmax + aggregation + residual (one wave per (head,node))
    gat_aggregate<<<(nPairs + 7) / 8, 256, 0, stream>>>(
        h, norm, z, sS, sD, src, offsets, eid, out, nNodes);

    // 5) second tuple output: e passthrough
    hipMemcpyAsync(out + (size_t)nNodes * INDIM, e, (size_t)nE * sizeof(float),
                   hipMemcpyDeviceToDevice, stream);
}